// ElementwiseSparsity_22883585753307
// MI455X (gfx1250) — compile-verified
//
#include <hip/hip_runtime.h>
#include <hip/hip_bf16.h>
#include <stdint.h>

#define BATCH 4
#define CDIM 1024   // MODEL_DIM (K of expand GEMM)
#define DDIM 2048   // HIGH_DIM  (M of expand GEMM)
#define TDIM 8192   // T         (N of expand GEMM)
#define KEEP 512

typedef __attribute__((ext_vector_type(2))) float v2f;
typedef __attribute__((ext_vector_type(8))) float v8f;
typedef int v4i_ __attribute__((vector_size(16)));

// ---------------- workspace layout (in uint32 units) ----------------
#define CAP     131072                      // candidate cap per batch
#define HIST_U  0                           // B*4096 histogram
#define CTRL_U  (BATCH*4096)                // B*8 control words
// ctrl[b]: 0=bucket 1=countAbove 2=threshKey 3=keepEq 4=tieCnt 5=nzCnt 6=candCnt
#define CAND_U  (CTRL_U + 64)               // B*CAP candidate low-keys
#define NZI_U   (CAND_U + BATCH*CAP)        // B*KEEP*2 (d,t)
#define NZV_U   (NZI_U + BATCH*KEEP*2)      // B*KEEP values

__device__ __forceinline__ unsigned monokey(float f) {
  unsigned u = __float_as_uint(f);
  unsigned mask = (unsigned)((int)u >> 31) | 0x80000000u;
  return u ^ mask;   // larger float -> larger key
}

// ---------------- 1) expand GEMM, fp32 WMMA, async-LDS double buffer ----------------
#define MT 128
#define NT 64
#define KC 32
#define APAD 4
#define BPAD 4

#if __has_builtin(__builtin_amdgcn_global_load_async_to_lds_b128)
#define HAVE_ASYNC 1
#else
#define HAVE_ASYNC 0
#endif

#if HAVE_ASYNC
__device__ __forceinline__ void async_cp16(const float* g, float* l) {
  __builtin_amdgcn_global_load_async_to_lds_b128(
      (__attribute__((address_space(1))) v4i_*)(float*)g,
      (__attribute__((address_space(3))) v4i_*)l,
      0, 0);
}
#if __has_builtin(__builtin_amdgcn_s_wait_asynccnt)
#define WAIT_ASYNC(N) __builtin_amdgcn_s_wait_asynccnt(N)
#else
#define WAIT_ASYNC(N) asm volatile("s_wait_asynccnt %0" :: "i"(N) : "memory")
#endif
#endif

__global__ __launch_bounds__(256) void expand_wmma(
    const float* __restrict__ x, const float* __restrict__ Wexp,
    const float* __restrict__ bexp, float* __restrict__ h)
{
  __shared__ float As[2][MT][KC + APAD];   // W_exp tiles (M x K), double buffered
  __shared__ float Bs[2][KC][NT + BPAD];   // x tiles     (K x N), double buffered
  const int b   = blockIdx.z;
  const int d0  = blockIdx.y * MT;
  const int t0  = blockIdx.x * NT;
  const int tid = threadIdx.x;
  const int lane = tid & 31;
  const int wave = tid >> 5;

  const float* xb = x + (size_t)b * CDIM * TDIM;

  // 8 waves in a 4(M) x 2(N) grid; each wave owns a 32x32 tile (2x2 WMMA subtiles)
  const int wm = (wave & 3) * 32;
  const int wn = (wave >> 2) * 32;

  const int mrow = lane & 15;        // M row (A) / N col (B,D)
  const int kb   = (lane >> 4) * 2;  // K pair selector for f32 frags

  v8f c00 = {}, c01 = {}, c10 = {}, c11 = {};

  // staging coordinates: A = 128x32 (16 floats/thread), B = 32x64 (8 floats/thread)
  const int la = tid * 16;
  const int ar = la / KC, ac = la % KC;   // ar 0..127, ac in {0,16}
  const int lb = tid * 8;
  const int br = lb / NT, bc = lb % NT;   // br 0..31,  bc multiple of 8

  auto stage = [&](int buf, int k0) {
#if HAVE_ASYNC
    const float* ga = Wexp + (size_t)(d0 + ar) * CDIM + k0 + ac;
    async_cp16(ga,      &As[buf][ar][ac]);
    async_cp16(ga + 4,  &As[buf][ar][ac + 4]);
    async_cp16(ga + 8,  &As[buf][ar][ac + 8]);
    async_cp16(ga + 12, &As[buf][ar][ac + 12]);
    const float* gb = xb + (size_t)(k0 + br) * TDIM + t0 + bc;
    async_cp16(gb,     &Bs[buf][br][bc]);
    async_cp16(gb + 4, &Bs[buf][br][bc + 4]);
#else
    const float4* sa = (const float4*)(Wexp + (size_t)(d0 + ar) * CDIM + k0 + ac);
    *(float4*)&As[buf][ar][ac]      = sa[0];
    *(float4*)&As[buf][ar][ac + 4]  = sa[1];
    *(float4*)&As[buf][ar][ac + 8]  = sa[2];
    *(float4*)&As[buf][ar][ac + 12] = sa[3];
    const float4* sb = (const float4*)(xb + (size_t)(k0 + br) * TDIM + t0 + bc);
    *(float4*)&Bs[buf][br][bc]     = sb[0];
    *(float4*)&Bs[buf][br][bc + 4] = sb[1];
#endif
  };

  stage(0, 0);
  for (int k0 = 0; k0 < CDIM; k0 += KC) {
    const int cur = (k0 / KC) & 1;
    if (k0 + KC < CDIM) {
      stage(cur ^ 1, k0 + KC);       // DMA next tile while computing current
#if HAVE_ASYNC
      WAIT_ASYNC(6);                 // 6 newer ops pending => older 6 retired (in-order)
#endif
    } else {
#if HAVE_ASYNC
      WAIT_ASYNC(0);
#endif
    }
    __syncthreads();

    #pragma unroll
    for (int kk = 0; kk < KC; kk += 4) {
      v2f a0, a1, b0, b1;
      a0.x = As[cur][wm + mrow][kk + kb];
      a0.y = As[cur][wm + mrow][kk + kb + 1];
      a1.x = As[cur][wm + 16 + mrow][kk + kb];
      a1.y = As[cur][wm + 16 + mrow][kk + kb + 1];
      b0.x = Bs[cur][kk + kb][wn + mrow];
      b0.y = Bs[cur][kk + kb + 1][wn + mrow];
      b1.x = Bs[cur][kk + kb][wn + 16 + mrow];
      b1.y = Bs[cur][kk + kb + 1][wn + 16 + mrow];
      c00 = __builtin_amdgcn_wmma_f32_16x16x4_f32(false, a0, false, b0, (short)0, c00, false, false);
      c01 = __builtin_amdgcn_wmma_f32_16x16x4_f32(false, a0, false, b1, (short)0, c01, false, false);
      c10 = __builtin_amdgcn_wmma_f32_16x16x4_f32(false, a1, false, b0, (short)0, c10, false, false);
      c11 = __builtin_amdgcn_wmma_f32_16x16x4_f32(false, a1, false, b1, (short)0, c11, false, false);
    }
    __syncthreads();
  }

  // epilogue: bias + store (D layout: VGPR r -> M=r / r+8 by lane half, N=lane&15)
  float* hb = h + (size_t)b * DDIM * TDIM;
  const int n0 = t0 + wn + mrow;
  #pragma unroll
  for (int r = 0; r < 8; ++r) {
    const int m = r + (lane >> 4) * 8;
    const int row0 = d0 + wm + m;
    const int row1 = d0 + wm + 16 + m;
    const float bi0 = bexp[row0];
    const float bi1 = bexp[row1];
    hb[(size_t)row0 * TDIM + n0]      = c00[r] + bi0;
    hb[(size_t)row0 * TDIM + n0 + 16] = c01[r] + bi0;
    hb[(size_t)row1 * TDIM + n0]      = c10[r] + bi1;
    hb[(size_t)row1 * TDIM + n0 + 16] = c11[r] + bi1;
  }
}

// ---------------- 2) top-k machinery ----------------
__global__ void zero_k(unsigned* __restrict__ ws) {
  int i = blockIdx.x * 256 + threadIdx.x;
  if (i < CTRL_U + 64) ws[i] = 0;
}

__global__ __launch_bounds__(256) void hist_k(const float* __restrict__ h,
                                              unsigned* __restrict__ ws) {
  __shared__ unsigned sh[4096];
  const int b = blockIdx.y;
  for (int i = threadIdx.x; i < 4096; i += 256) sh[i] = 0;
  __syncthreads();
  const float* hb = h + (size_t)b * DDIM * TDIM;
  const int total = DDIM * TDIM;
  for (int i = blockIdx.x * 256 + threadIdx.x; i < total; i += gridDim.x * 256)
    atomicAdd(&sh[monokey(hb[i]) >> 20], 1u);
  __syncthreads();
  unsigned* hist = ws + HIST_U + b * 4096;
  for (int i = threadIdx.x; i < 4096; i += 256)
    if (sh[i]) atomicAdd(&hist[i], sh[i]);
}

__global__ void scan_k(unsigned* __restrict__ ws) {
  const int b = blockIdx.x;
  const unsigned* hist = ws + HIST_U + b * 4096;
  unsigned* ctrl = ws + CTRL_U + b * 8;
  unsigned cum = 0;
  for (int bkt = 4095; bkt >= 0; --bkt) {
    unsigned c = hist[bkt];
    if (cum + c >= KEEP) { ctrl[0] = (unsigned)bkt; ctrl[1] = cum; return; }
    cum += c;
  }
  ctrl[0] = 0; ctrl[1] = cum;
}

__global__ __launch_bounds__(256) void collect_k(const float* __restrict__ h,
                                                 unsigned* __restrict__ ws) {
  const int b = blockIdx.y;
  unsigned* ctrl = ws + CTRL_U + b * 8;
  unsigned* cand = ws + CAND_U + (size_t)b * CAP;
  const unsigned bucket = ctrl[0];
  const float* hb = h + (size_t)b * DDIM * TDIM;
  const int total = DDIM * TDIM;
  for (int i = blockIdx.x * 256 + threadIdx.x; i < total; i += gridDim.x * 256) {
    unsigned key = monokey(hb[i]);
    if ((key >> 20) == bucket) {
      unsigned p = atomicAdd(&ctrl[6], 1u);
      if (p < CAP) cand[p] = key & 0xFFFFFu;
    }
  }
}

__global__ __launch_bounds__(256) void select_k(unsigned* __restrict__ ws) {
  const int b = blockIdx.x;
  unsigned* ctrl = ws + CTRL_U + b * 8;
  const unsigned* cand = ws + CAND_U + (size_t)b * CAP;
  __shared__ unsigned scnt;
  const unsigned n = ctrl[6] < (unsigned)CAP ? ctrl[6] : (unsigned)CAP;
  const unsigned rem = KEEP - ctrl[1];
  unsigned cur = 0;
  for (int bit = 19; bit >= 0; --bit) {
    unsigned trial = cur | (1u << bit);
    if (threadIdx.x == 0) scnt = 0;
    __syncthreads();
    unsigned local = 0;
    for (unsigned i = threadIdx.x; i < n; i += 256) if (cand[i] >= trial) ++local;
    if (local) atomicAdd(&scnt, local);
    __syncthreads();
    unsigned tot = scnt;
    __syncthreads();
    if (tot >= rem) cur = trial;
  }
  if (threadIdx.x == 0) scnt = 0;
  __syncthreads();
  unsigned local = 0;
  for (unsigned i = threadIdx.x; i < n; i += 256) if (cand[i] > cur) ++local;
  if (local) atomicAdd(&scnt, local);
  __syncthreads();
  if (threadIdx.x == 0) {
    unsigned above = ctrl[1] + scnt;            // strictly greater than thresh
    ctrl[2] = (ctrl[0] << 20) | cur;            // exact Kth key
    ctrl[3] = KEEP - above;                     // tie budget (>=1)
  }
}

// ---------------- 3) sparsify in place + compact nonzeros ----------------
__global__ __launch_bounds__(256) void finalize_k(float* __restrict__ h,
                                                  unsigned* __restrict__ ws) {
  const size_t total = (size_t)BATCH * DDIM * TDIM;
  size_t i = (size_t)blockIdx.x * 256 + threadIdx.x;
  if (i >= total) return;
  const int b = (int)(i / ((size_t)DDIM * TDIM));
  const int j = (int)(i % ((size_t)DDIM * TDIM));
  unsigned* ctrl = ws + CTRL_U + b * 8;
  float v = h[i];
  unsigned key = monokey(v);
  unsigned tk = ctrl[2];
  bool keep = false;
  if (key > tk) keep = true;
  else if (key == tk) {
    unsigned p = atomicAdd(&ctrl[4], 1u);
    keep = (p < ctrl[3]);
  }
  if (keep) {
    unsigned p = atomicAdd(&ctrl[5], 1u);
    if (p < KEEP) {
      unsigned* nzi = ws + NZI_U + (size_t)b * KEEP * 2;
      float* nzv = (float*)(ws + NZV_U) + (size_t)b * KEEP;
      nzi[p * 2]     = (unsigned)(j / TDIM);   // d
      nzi[p * 2 + 1] = (unsigned)(j % TDIM);   // t
      nzv[p] = v;
    }
  } else {
    h[i] = 0.0f;
  }
}

// ---------------- 4) contract: bias fill + sparse rank-1 scatter ----------------
__global__ __launch_bounds__(256) void fill_out_k(float* __restrict__ out,
                                                  const float* __restrict__ bcon) {
  const size_t total = (size_t)BATCH * CDIM * TDIM;
  size_t i = (size_t)blockIdx.x * 256 + threadIdx.x;
  if (i >= total) return;
  int c = (int)((i / TDIM) % CDIM);
  out[i] = bcon[c];
}

__global__ __launch_bounds__(256) void scatter_k(float* __restrict__ out,
                                                 const float* __restrict__ Wcon,
                                                 const unsigned* __restrict__ ws) {
  const int b = blockIdx.y;
  const unsigned* ctrl = ws + CTRL_U + b * 8;
  unsigned nz = ctrl[5] < (unsigned)KEEP ? ctrl[5] : (unsigned)KEEP;
  unsigned i = blockIdx.x;
  if (i >= nz) return;
  const unsigned* nzi = ws + NZI_U + (size_t)b * KEEP * 2;
  const float* nzv = (const float*)(ws + NZV_U) + (size_t)b * KEEP;
  const unsigned d = nzi[i * 2];
  const unsigned t = nzi[i * 2 + 1];
  const float v = nzv[i];
  float* ob = out + (size_t)b * CDIM * TDIM;
  for (int c = threadIdx.x; c < CDIM; c += 256)
    unsafeAtomicAdd(&ob[(size_t)c * TDIM + t], v * Wcon[(size_t)c * DDIM + d]);
}

// ---------------- launch ----------------
extern "C" void kernel_launch(void* const* d_in, const int* in_sizes, int n_in,
                              void* d_out, int out_size, void* d_ws, size_t ws_size,
                              hipStream_t stream) {
  const float* x    = (const float*)d_in[0];
  const float* Wexp = (const float*)d_in[1];
  const float* bexp = (const float*)d_in[2];
  const float* Wcon = (const float*)d_in[3];
  const float* bcon = (const float*)d_in[4];
  float* out    = (float*)d_out;                        // [B, MODEL, T]
  float* sparse = out + (size_t)BATCH * CDIM * TDIM;    // [B, HIGH, T], doubles as h
  unsigned* ws = (unsigned*)d_ws;

  zero_k<<<(CTRL_U + 64 + 255) / 256, 256, 0, stream>>>(ws);

  dim3 gg(TDIM / NT, DDIM / MT, BATCH);
  expand_wmma<<<gg, 256, 0, stream>>>(x, Wexp, bexp, sparse);

  hist_k<<<dim3(1024, BATCH), 256, 0, stream>>>(sparse, ws);
  scan_k<<<BATCH, 1, 0, stream>>>(ws);
  collect_k<<<dim3(1024, BATCH), 256, 0, stream>>>(sparse, ws);
  select_k<<<BATCH, 256, 0, stream>>>(ws);

  size_t totS = (size_t)BATCH * DDIM * TDIM;
  finalize_k<<<(unsigned)((totS + 255) / 256), 256, 0, stream>>>(sparse, ws);

  size_t totO = (size_t)BATCH * CDIM * TDIM;
  fill_out_k<<<(unsigned)((totO + 255) / 256), 256, 0, stream>>>(out, bcon);
  scatter_k<<<dim3(KEEP, BATCH), 256, 0, stream>>>(out, Wcon, (const unsigned*)ws);
}